// TSQBEVCore_71236327572185
// MI455X (gfx1250) — compile-verified
//
#include <hip/hip_runtime.h>
#include <hip/hip_bf16.h>

typedef __attribute__((ext_vector_type(2))) float v2f;
typedef __attribute__((ext_vector_type(8))) float v8f;

#define B_   2
#define V_   6
#define C_   256
#define Q_   900
#define KP_  4
#define H8_  56
#define W8_  100
#define H16_ 28
#define W16_ 50
#define HW8_  (H8_*W8_)
#define HW16_ (H16_*W16_)

// ---------------------------------------------------------------------------
// CHW -> HWC layout transform (per (b,v) plane), 32x32 LDS tile transpose.
// ---------------------------------------------------------------------------
__global__ __launch_bounds__(256) void transpose_chw_hwc(
    const float* __restrict__ src, float* __restrict__ dst, int HW) {
  __shared__ float tile[32][33];
  const int bv = blockIdx.z;
  const int c0 = blockIdx.y * 32;
  const int p0 = blockIdx.x * 32;
  const int tx = threadIdx.x & 31;
  const int ty = threadIdx.x >> 5;           // 0..7
  const float* s = src + (size_t)bv * C_ * HW;
  float*       d = dst + (size_t)bv * HW * C_;
#pragma unroll
  for (int i = 0; i < 4; ++i) {
    const int cl = ty + 8 * i;
    const int pp = p0 + tx;
    if (pp < HW) tile[cl][tx] = s[(size_t)(c0 + cl) * HW + pp];
  }
  __syncthreads();
#pragma unroll
  for (int i = 0; i < 4; ++i) {
    const int pp = p0 + ty + 8 * i;
    if (pp < HW) d[(size_t)pp * C_ + (c0 + tx)] = tile[tx][ty + 8 * i];
  }
}

// ---------------------------------------------------------------------------
// Fused projection (WMMA f32 16x16x4) + masked bilinear sampling.
// One block = (batch b, 4 queries).
// Phase 1: waves 0..5 each project 16 points (4q x 4kp) for one view:
//          D[m][n] = sum_k pt[m][k] * P[n][k], P = K @ E[:3,:4].
// Phase 2 (TRANSPOSED): thread t -> query tid>>6, channels 4*(tid&63)..+3;
//          each bilinear corner is one global_load_b128 (HWC layout).
// Phase 2 (fallback):   thread t -> channel t, scalar strided gathers.
// ---------------------------------------------------------------------------
template <bool TRANSPOSED>
__global__ __launch_bounds__(256) void sample_fuse(
    const float* __restrict__ l0, const float* __restrict__ l1,
    const float* __restrict__ refs, const float* __restrict__ intr,
    const float* __restrict__ extr, const float* __restrict__ keyp,
    float* __restrict__ out) {
  __shared__ float su[96], sv[96], sz[96];   // [view][4q*4kp]

  const int tid  = threadIdx.x;
  const int lane = tid & 31;
  const int wave = tid >> 5;
  const int b    = blockIdx.x / 225;
  const int q0   = (blockIdx.x % 225) * 4;

  // ---- Phase 1: projection via WMMA (waves 0..5, one view each) ----
  if (wave < 6) {
    const int v = wave;
    const float* Km = intr + (size_t)(b * V_ + v) * 9;
    const float* Em = extr + (size_t)(b * V_ + v) * 16;
    float P[3][4];                            // P = K @ E[:3,:4]
#pragma unroll
    for (int i = 0; i < 3; ++i)
#pragma unroll
      for (int j = 0; j < 4; ++j)
        P[i][j] = Km[i * 3 + 0] * Em[0 * 4 + j] +
                  Km[i * 3 + 1] * Em[1 * 4 + j] +
                  Km[i * 3 + 2] * Em[2 * 4 + j];

    const int m  = lane & 15;                 // point index 0..15
    const int kh = lane >> 4;                 // K-half: 0 -> K=0,1 ; 1 -> K=2,3
    const int ql = m >> 2, kp = m & 3;
    const int q  = q0 + ql;
    const float px = refs[((size_t)b * Q_ + q) * 3 + 0] + keyp[kp * 3 + 0];
    const float py = refs[((size_t)b * Q_ + q) * 3 + 1] + keyp[kp * 3 + 1];
    const float pz = refs[((size_t)b * Q_ + q) * 3 + 2] + keyp[kp * 3 + 2];

    v2f a;                                    // A 16x4: [x y z 1]
    a.x = kh ? pz : px;
    a.y = kh ? 1.0f : py;

    const int n = m;                          // output column
    // Row n of P selected via cndmask chains (no dynamic index -> no spill).
    float Pn[4];
#pragma unroll
    for (int j = 0; j < 4; ++j)
      Pn[j] = (n == 0) ? P[0][j] : (n == 1) ? P[1][j]
            : (n == 2) ? P[2][j] : 0.0f;
    v2f bb;                                   // B[k][n] = P[n][k]
    bb.x = kh ? Pn[2] : Pn[0];
    bb.y = kh ? Pn[3] : Pn[1];

    v8f c = {};
    c = __builtin_amdgcn_wmma_f32_16x16x4_f32(
        false, a, false, bb, (short)0, c, false, false);

    if (n < 3) {                              // lanes 0,1,2 / 16,17,18 scatter
      float* dstp = (n == 0) ? su : (n == 1) ? sv : sz;
#pragma unroll
      for (int r = 0; r < 8; ++r)             // VGPR r -> row M = r + 8*kh
        dstp[v * 16 + r + 8 * kh] = c[r];
    }
  }
  __syncthreads();
  if (tid < 96) {                             // u = u'/max(z,eps); v likewise
    const float inv = 1.0f / fmaxf(sz[tid], 1e-5f);
    su[tid] *= inv;
    sv[tid] *= inv;
  }
  __syncthreads();

  // ---- Phase 2 ----
  if constexpr (TRANSPOSED) {
    const int ql = tid >> 6;                  // 0..3: one query per thread
    const int cc = (tid & 63) * 4;            // 4 consecutive channels
    const int q  = q0 + ql;

    float cnt = 0.0f;
#pragma unroll
    for (int v = 0; v < V_; ++v)
#pragma unroll
      for (int kp = 0; kp < KP_; ++kp)
        cnt += (sz[v * 16 + ql * 4 + kp] > 0.0f) ? 1.0f : 0.0f;

    float4 acc = make_float4(0.f, 0.f, 0.f, 0.f);
#pragma unroll 1
    for (int lvl = 0; lvl < 2; ++lvl) {
      const float* fm = lvl ? l1 : l0;
      const int H  = lvl ? H16_ : H8_;
      const int W  = lvl ? W16_ : W8_;
      const int HW = H * W;
#pragma unroll 1
      for (int v = 0; v < V_; ++v) {
        const float* basep = fm + (size_t)(b * V_ + v) * HW * C_ + cc;
#pragma unroll
        for (int kp = 0; kp < KP_; ++kp) {
          const int i = v * 16 + ql * 4 + kp;
          const float z = sz[i];
          if (z > 0.0f) {
            const float x = fminf(fmaxf(su[i], -1e8f), 1e8f);
            const float y = fminf(fmaxf(sv[i], -1e8f), 1e8f);
            const float xf = floorf(x), yf = floorf(y);
            const int x0 = (int)xf, y0 = (int)yf;
            const int x1 = x0 + 1,  y1 = y0 + 1;
            const float wx = x - xf, wy = y - yf;
            const float mx0 = (x0 >= 0 && x0 < W) ? 1.0f : 0.0f;
            const float mx1 = (x1 >= 0 && x1 < W) ? 1.0f : 0.0f;
            const float my0 = (y0 >= 0 && y0 < H) ? 1.0f : 0.0f;
            const float my1 = (y1 >= 0 && y1 < H) ? 1.0f : 0.0f;
            const int xc0 = min(max(x0, 0), W - 1);
            const int xc1 = min(max(x1, 0), W - 1);
            const int yc0 = min(max(y0, 0), H - 1);
            const int yc1 = min(max(y1, 0), H - 1);
            const float w00 = mx0 * my0 * (1.0f - wx) * (1.0f - wy);
            const float w01 = mx1 * my0 * wx * (1.0f - wy);
            const float w10 = mx0 * my1 * (1.0f - wx) * wy;
            const float w11 = mx1 * my1 * wx * wy;
            const float4 g00 = *(const float4*)(basep + (size_t)(yc0 * W + xc0) * C_);
            const float4 g01 = *(const float4*)(basep + (size_t)(yc0 * W + xc1) * C_);
            const float4 g10 = *(const float4*)(basep + (size_t)(yc1 * W + xc0) * C_);
            const float4 g11 = *(const float4*)(basep + (size_t)(yc1 * W + xc1) * C_);
            acc.x += g00.x * w00 + g01.x * w01 + g10.x * w10 + g11.x * w11;
            acc.y += g00.y * w00 + g01.y * w01 + g10.y * w10 + g11.y * w11;
            acc.z += g00.z * w00 + g01.z * w01 + g10.z * w10 + g11.z * w11;
            acc.w += g00.w * w00 + g01.w * w01 + g10.w * w10 + g11.w * w11;
          }
        }
      }
    }
    const float s = 1.0f / (2.0f * fmaxf(cnt, 1.0f));
    float4 r = make_float4(acc.x * s, acc.y * s, acc.z * s, acc.w * s);
    *(float4*)(out + ((size_t)b * Q_ + q) * C_ + cc) = r;
  } else {
    // Fallback: original [C,H,W] layout, scalar strided gathers.
    const int c = tid;
#pragma unroll 1
    for (int ql = 0; ql < 4; ++ql) {
      const int q = q0 + ql;
      float cnt = 0.0f;
#pragma unroll
      for (int v = 0; v < V_; ++v)
#pragma unroll
        for (int kp = 0; kp < KP_; ++kp)
          cnt += (sz[v * 16 + ql * 4 + kp] > 0.0f) ? 1.0f : 0.0f;

      float acc = 0.0f;
#pragma unroll 1
      for (int lvl = 0; lvl < 2; ++lvl) {
        const float* fm = lvl ? l1 : l0;
        const int H  = lvl ? H16_ : H8_;
        const int W  = lvl ? W16_ : W8_;
        const int HW = H * W;
#pragma unroll 1
        for (int v = 0; v < V_; ++v) {
          const float* basep = fm + ((size_t)(b * V_ + v) * C_ + c) * HW;
#pragma unroll
          for (int kp = 0; kp < KP_; ++kp) {
            const int i = v * 16 + ql * 4 + kp;
            const float z = sz[i];
            if (z > 0.0f) {
              const float x = fminf(fmaxf(su[i], -1e8f), 1e8f);
              const float y = fminf(fmaxf(sv[i], -1e8f), 1e8f);
              const float xf = floorf(x), yf = floorf(y);
              const int x0 = (int)xf, y0 = (int)yf;
              const int x1 = x0 + 1,  y1 = y0 + 1;
              const float wx = x - xf, wy = y - yf;
              const float mx0 = (x0 >= 0 && x0 < W) ? 1.0f : 0.0f;
              const float mx1 = (x1 >= 0 && x1 < W) ? 1.0f : 0.0f;
              const float my0 = (y0 >= 0 && y0 < H) ? 1.0f : 0.0f;
              const float my1 = (y1 >= 0 && y1 < H) ? 1.0f : 0.0f;
              const int xc0 = min(max(x0, 0), W - 1);
              const int xc1 = min(max(x1, 0), W - 1);
              const int yc0 = min(max(y0, 0), H - 1);
              const int yc1 = min(max(y1, 0), H - 1);
              const float g00 = fm == nullptr ? 0.f : basep[yc0 * W + xc0];
              const float g01 = basep[yc0 * W + xc1];
              const float g10 = basep[yc1 * W + xc0];
              const float g11 = basep[yc1 * W + xc1];
              acc += (g00 * mx0 * my0) * ((1.0f - wx) * (1.0f - wy)) +
                     (g01 * mx1 * my0) * (wx * (1.0f - wy)) +
                     (g10 * mx0 * my1) * ((1.0f - wx) * wy) +
                     (g11 * mx1 * my1) * (wx * wy);
            }
          }
        }
      }
      out[((size_t)b * Q_ + q) * C_ + c] = acc / (2.0f * fmaxf(cnt, 1.0f));
    }
  }
}

// ---------------------------------------------------------------------------
extern "C" void kernel_launch(void* const* d_in, const int* in_sizes, int n_in,
                              void* d_out, int out_size, void* d_ws,
                              size_t ws_size, hipStream_t stream) {
  const float* f8   = (const float*)d_in[0];
  const float* f16  = (const float*)d_in[1];
  const float* refs = (const float*)d_in[2];
  const float* intr = (const float*)d_in[3];
  const float* extr = (const float*)d_in[4];
  const float* keyp = (const float*)d_in[5];
  float* out = (float*)d_out;

  const size_t n8  = (size_t)B_ * V_ * C_ * HW8_;   // 17,203,200
  const size_t n16 = (size_t)B_ * V_ * C_ * HW16_;  //  4,300,800
  const size_t need = (n8 + n16) * sizeof(float);   // ~86 MB

  const dim3 blk(256);
  const dim3 grid_main(B_ * (Q_ / 4));              // 450 blocks

  if (ws_size >= need) {
    float* t8  = (float*)d_ws;
    float* t16 = t8 + n8;
    const dim3 g8(HW8_ / 32, C_ / 32, B_ * V_);           // (175, 8, 12)
    const dim3 g16((HW16_ + 31) / 32, C_ / 32, B_ * V_);  // ( 44, 8, 12)
    transpose_chw_hwc<<<g8,  blk, 0, stream>>>(f8,  t8,  HW8_);
    transpose_chw_hwc<<<g16, blk, 0, stream>>>(f16, t16, HW16_);
    sample_fuse<true><<<grid_main, blk, 0, stream>>>(t8, t16, refs, intr,
                                                     extr, keyp, out);
  } else {
    sample_fuse<false><<<grid_main, blk, 0, stream>>>(f8, f16, refs, intr,
                                                      extr, keyp, out);
  }
}